// CmxuLayer_22376779612863
// MI455X (gfx1250) — compile-verified
//
#include <hip/hip_runtime.h>

typedef float v2f __attribute__((ext_vector_type(2)));
typedef float v8f __attribute__((ext_vector_type(8)));

#define NCH 6

// K-slot permutation: slot s of the 16-wide K dimension carries x-row
// slot_src[s] (0..5 = field_re row, 6..11 = field_im row-6). Chosen so each
// WMMA B-fragment VGPR {slot k, slot k+2} reads from ONE source array:
//   j=0 slots{0,2}=re0,re2   j=1 slots{1,3}=re1,re3
//   j=2 slots{4,6}=re4,re5   j=3 slots{5,7}=im0,im1
//   j=4 slots{8,10}=im2,im3  j=5 slots{9,11}=im4,im5
__constant__ const int kSlotSrc[12] = {0, 1, 2, 3, 4, 6, 5, 7, 8, 10, 9, 11};

// ---------------------------------------------------------------------------
// Kernel 1: replay the sequential MZI mesh (15 Givens rotations + output
// phases) exactly as the reference, then emit the real-ified matrix M
// (16x16 fp32 row-major) into workspace:
//   D-row c   (c=0..5): out_re channel c coefficients
//   D-row 8+c (c=0..5): out_im channel c coefficients
// Rows 6,7,14,15 and slots 12..15 are zero, so the apply kernel's D VGPRs
// 0..5 are valid for all 32 lanes and VGPRs 6..7 are uniformly dead.
// ---------------------------------------------------------------------------
__global__ void cmxu_build_unitary_kernel(const float* __restrict__ mzi,
                                          const float* __restrict__ oph,
                                          float* __restrict__ Mws) {
    if (threadIdx.x != 0 || blockIdx.x != 0) return;
    float a[NCH][NCH], b[NCH][NCH];
    for (int r = 0; r < NCH; ++r)
        for (int c = 0; c < NCH; ++c) { a[r][c] = (r == c) ? 1.f : 0.f; b[r][c] = 0.f; }
    int idx = 0;
    for (int i = 0; i < NCH; ++i) {
        for (int j = i + 1; j < NCH; ++j) {
            float th = mzi[idx], ph = mzi[idx + 1]; idx += 2;
            float ct = cosf(th), st = sinf(th);
            float cp = cosf(ph), sp = sinf(ph);
            for (int col = 0; col < NCH; ++col) {
                float air = a[i][col], aii = b[i][col];
                float ajr = a[j][col], aji = b[j][col];
                // row_i = exp(i*phi)*cos(th)*U[i] + sin(th)*U[j]
                float ri_r = (ct*cp)*air - (ct*sp)*aii + st*ajr;
                float ri_i = (ct*cp)*aii + (ct*sp)*air + st*aji;
                // row_j = -exp(i*phi)*sin(th)*U[i] + cos(th)*U[j]
                float rj_r = -((st*cp)*air - (st*sp)*aii) + ct*ajr;
                float rj_i = -((st*cp)*aii + (st*sp)*air) + ct*aji;
                a[i][col] = ri_r; b[i][col] = ri_i;
                a[j][col] = rj_r; b[j][col] = rj_i;
            }
        }
    }
    for (int r = 0; r < NCH; ++r) {
        float co = cosf(oph[r]), so = sinf(oph[r]);
        for (int c = 0; c < NCH; ++c) {
            float ar = a[r][c], ai = b[r][c];
            a[r][c] = ar*co - ai*so;
            b[r][c] = ar*so + ai*co;
        }
    }
    // out[c] = sum_k U[k][c] * x[k]:
    //   y_re[c] = sum_k a[k][c]*xre[k] - b[k][c]*xim[k]
    //   y_im[c] = sum_k b[k][c]*xre[k] + a[k][c]*xim[k]
    for (int i = 0; i < 256; ++i) Mws[i] = 0.f;
    for (int c = 0; c < NCH; ++c) {
        for (int s = 0; s < 12; ++s) {
            int r = kSlotSrc[s];
            if (r < 6) {            // slot carries field_re row r
                Mws[c*16 + s]       =  a[r][c];
                Mws[(8 + c)*16 + s] =  b[r][c];
            } else {                // slot carries field_im row r-6
                int ri = r - 6;
                Mws[c*16 + s]       = -b[ri][c];
                Mws[(8 + c)*16 + s] =  a[ri][c];
            }
        }
    }
}

// ---------------------------------------------------------------------------
// Kernel 2: stream 201 MB through chained V_WMMA_F32_16X16X4_F32 (K=12 in 3
// chunks), 2 tiles (32 samples) per iteration. All loads/stores are
// SADDR + 32-bit-voffset(scale_offset) form; the second tile reuses the same
// index VGPRs via constant IOFFSET (+64B loads / +128B stores). Channel
// output bases arrive as six independent kernel args so every store shares
// one index VGPR. Scalar (readfirstlane) loop => EXEC all-ones for WMMA.
// ---------------------------------------------------------------------------
__global__ void __launch_bounds__(256)
cmxu_apply_kernel(const float* __restrict__ fre,
                  const float* __restrict__ fim,
                  const float* __restrict__ Mws,
                  float* __restrict__ o0, float* __restrict__ o1,
                  float* __restrict__ o2, float* __restrict__ o3,
                  float* __restrict__ o4, float* __restrict__ o5,
                  int batch, int npairs) {
    const int lane = threadIdx.x & 31;
    const unsigned n    = (unsigned)(lane & 15);  // B/D column (sample in tile)
    const unsigned half = (unsigned)(lane >> 4);  // 0: lanes 0-15, 1: 16-31
    const int wid = __builtin_amdgcn_readfirstlane(
        blockIdx.x * (blockDim.x >> 5) + (threadIdx.x >> 5));
    const int nw  = __builtin_amdgcn_readfirstlane(gridDim.x * (blockDim.x >> 5));
    const unsigned ub = (unsigned)batch;

    // A-matrix fragments (16x4 fp32): lane = row (lane&15), VGPR r holds
    // k-slot r + 2*half within the chunk.
    v2f A0, A1, A2;
    A0.x = Mws[n*16 + 0 + 2*half];  A0.y = Mws[n*16 + 1 + 2*half];
    A1.x = Mws[n*16 + 4 + 2*half];  A1.y = Mws[n*16 + 5 + 2*half];
    A2.x = Mws[n*16 + 8 + 2*half];  A2.y = Mws[n*16 + 9 + 2*half];

    // Per-lane 32-bit row offsets (elements) from UNIFORM bases fre/fim —
    // the K-slot permutation guarantees one source array per load.
    const unsigned ro0 = (0u + 2u*half) * ub + n;  // fre rows 0 / 2
    const unsigned ro1 = (1u + 2u*half) * ub + n;  // fre rows 1 / 3
    const unsigned ro2 = (4u +     half) * ub + n; // fre rows 4 / 5
    const unsigned ro3 = (0u +     half) * ub + n; // fim rows 0 / 1
    const unsigned ro4 = (2u +     half) * ub + n; // fim rows 2 / 3
    const unsigned ro5 = (3u +     half) * ub + n; // fim rows 3 / 4

    // Store index: D VGPR v, this lane -> row v + 8*half:
    // half 0 = re ch v, half 1 = im ch v of the SAME 16 samples, so each
    // store instruction fills one 128B cacheline of interleaved complex64.
    const unsigned pl = half + 2u*n;

    for (int t = wid; t < npairs; t += nw) {
        const unsigned bt = (unsigned)t * 32u;   // first of two adjacent tiles

        v2f B0a, B1a, B2a, B0b, B1b, B2b;
        B0a.x = fre[ro0 + bt];        B0a.y = fre[ro1 + bt];
        B1a.x = fre[ro2 + bt];        B1a.y = fim[ro3 + bt];
        B2a.x = fim[ro4 + bt];        B2a.y = fim[ro5 + bt];
        B0b.x = fre[ro0 + bt + 16u];  B0b.y = fre[ro1 + bt + 16u];
        B1b.x = fre[ro2 + bt + 16u];  B1b.y = fim[ro3 + bt + 16u];
        B2b.x = fim[ro4 + bt + 16u];  B2b.y = fim[ro5 + bt + 16u];

        v8f accA = {}, accB = {};
        // 8 args: (neg_a, A, neg_b, B, c_mod, C, reuse_a, reuse_b)
        accA = __builtin_amdgcn_wmma_f32_16x16x4_f32(false, A0, false, B0a,
                                                     (short)0, accA, false, false);
        accB = __builtin_amdgcn_wmma_f32_16x16x4_f32(false, A0, false, B0b,
                                                     (short)0, accB, false, false);
        accA = __builtin_amdgcn_wmma_f32_16x16x4_f32(false, A1, false, B1a,
                                                     (short)0, accA, false, false);
        accB = __builtin_amdgcn_wmma_f32_16x16x4_f32(false, A1, false, B1b,
                                                     (short)0, accB, false, false);
        accA = __builtin_amdgcn_wmma_f32_16x16x4_f32(false, A2, false, B2a,
                                                     (short)0, accA, false, false);
        accB = __builtin_amdgcn_wmma_f32_16x16x4_f32(false, A2, false, B2b,
                                                     (short)0, accB, false, false);

        const unsigned u = 2u * bt + pl;
        o0[u] = accA[0];  o1[u] = accA[1];  o2[u] = accA[2];
        o3[u] = accA[3];  o4[u] = accA[4];  o5[u] = accA[5];
        o0[u + 32u] = accB[0];  o1[u + 32u] = accB[1];  o2[u + 32u] = accB[2];
        o3[u + 32u] = accB[3];  o4[u + 32u] = accB[4];  o5[u + 32u] = accB[5];
    }
}

extern "C" void kernel_launch(void* const* d_in, const int* in_sizes, int n_in,
                              void* d_out, int out_size, void* d_ws, size_t ws_size,
                              hipStream_t stream) {
    const float* fre = (const float*)d_in[0];
    const float* fim = (const float*)d_in[1];
    const float* mzi = (const float*)d_in[2];
    const float* oph = (const float*)d_in[3];
    float* out = (float*)d_out;          // complex64 viewed as interleaved fp32
    float* Mws = (float*)d_ws;           // 16x16 fp32 = 1 KB scratch

    const int batch  = in_sizes[0] / NCH;
    const int npairs = batch / 32;       // BATCH = 2^21 -> 65536 tile-pairs

    cmxu_build_unitary_kernel<<<1, 32, 0, stream>>>(mzi, oph, Mws);

    // Independent per-channel output bases -> SADDR-form stores in-kernel.
    float* o0 = out;
    float* o1 = out + 2 * (size_t)batch;
    float* o2 = out + 4 * (size_t)batch;
    float* o3 = out + 6 * (size_t)batch;
    float* o4 = out + 8 * (size_t)batch;
    float* o5 = out + 10 * (size_t)batch;

    // ~200 MB streamed; 8192 waves (8 tile-pairs each) saturate 23.3 TB/s.
    cmxu_apply_kernel<<<1024, 256, 0, stream>>>(fre, fim, Mws,
                                                o0, o1, o2, o3, o4, o5,
                                                batch, npairs);
}